// FARNN_60902636257583
// MI455X (gfx1250) — compile-verified
//
#include <hip/hip_runtime.h>

typedef __attribute__((ext_vector_type(2))) float v2f;
typedef __attribute__((ext_vector_type(8))) float v8f;

#define B_SZ   256
#define L_SEQ  512
#define D_DIM  300
#define R_DIM  150
#define SAS    200
#define RP     160   // R padded  -> 10 tiles of 16
#define SASP   208   // SAS padded-> 13 tiles of 16
#define HP     210   // LDS pitch for h  (even: b64-aligned pairs; 210%64=18, 16 rows conflict-free)
#define PP     162   // LDS pitch for P  (even; 162%64=34, conflict-free)

// packed-B tile geometry: ksteps of 4, ntiles of 16
#define G_KS   75    // D=300 /4
#define G_NT   10    // RP/16
#define T1_KS  50    // SAS=200 /4
#define T1_NT  10
#define T2_KS  40    // RP=160 /4
#define T2_NT  13    // SASP/16
#define W_KS   50
#define W_NT   13

static __device__ __forceinline__ v8f wmma_f32_k4(v2f a, v2f b, v8f c) {
  return __builtin_amdgcn_wmma_f32_16x16x4_f32(false, a, false, b,
                                               (short)0, c, false, false);
}

// ---------------------------------------------------------------------------
// Pack a K x N matrix into WMMA-B lane order, zero-padded.
// dst[((nt*ksteps + ks)*32 + lane)] = { B[4*ks+2*half][nt*16+l15],
//                                      B[4*ks+2*half+1][...] }
// element B[k][n] = src[k*strideK + n*strideN]  (strides allow transpose)
// ---------------------------------------------------------------------------
__global__ void __launch_bounds__(256)
pack_b(const float* __restrict__ src, int kBound, int nBound,
       int strideK, int strideN, int ksteps, int total, v2f* __restrict__ dst)
{
  const int id = blockIdx.x * 256 + threadIdx.x;
  if (id >= total) return;
  const int lane = id & 31;
  const int ks   = (id >> 5) % ksteps;
  const int nt   = (id >> 5) / ksteps;
  const int n = nt * 16 + (lane & 15);
  const int k = ks * 4 + 2 * (lane >> 4);
  v2f v;
  v.x = (k     < kBound && n < nBound) ? src[(long)k * strideK + (long)n * strideN] : 0.0f;
  v.y = (k + 1 < kBound && n < nBound) ? src[(long)(k + 1) * strideK + (long)n * strideN] : 0.0f;
  dst[id] = v;
}

// ---------------------------------------------------------------------------
// Kernel 1: L_all[m, r] = beta[r]*embed_r[input[m], r]
//                       + (1-beta[r])*relu( emb[input[m], :] @ G[:, r] )
// One wave per 16x16 tile. Inner loop: two b64 loads + one WMMA, no guards.
// ---------------------------------------------------------------------------
__global__ void __launch_bounds__(256)
farnn_embed_mix(const int* __restrict__ input,
                const float* __restrict__ embedding,
                const float* __restrict__ embed_r,
                const v2f*   __restrict__ Gp,     // packed [G_NT][G_KS][32]
                const float* __restrict__ beta,
                float* __restrict__ Lall)         // [B*L, RP]
{
  const int lane = threadIdx.x & 31;
  const int wv   = threadIdx.x >> 5;
  const int tile = blockIdx.x * 8 + wv;
  const int mTile = tile / G_NT;
  const int nTile = tile % G_NT;
  const int half = lane >> 4;
  const int l15  = lane & 15;

  const int  mrow  = mTile * 16 + l15;
  const long vocab = (long)input[mrow];
  const v2f* ap = (const v2f*)(embedding + vocab * D_DIM + 2 * half);
  const v2f* bp = Gp + (size_t)nTile * G_KS * 32 + lane;
  const int  n  = nTile * 16 + l15;

  v8f acc = {};
  for (int ks = 0; ks < G_KS; ++ks) {
    v2f a = ap[2 * ks];          // emb row: floats (4ks+2half, +1)
    v2f b = bp[ks * 32];         // packed, fully coalesced
    acc = wmma_f32_k4(a, b, acc);
  }

#pragma unroll
  for (int v = 0; v < 8; ++v) {
    const int m = mTile * 16 + v + 8 * half;
    float res = 0.0f;
    if (n < R_DIM) {
      const float lg = fmaxf(acc[v], 0.0f);
      const float er = embed_r[(long)input[m] * R_DIM + n];
      const float bv = beta[n];
      res = bv * er + (1.0f - bv) * lg;
    }
    Lall[(long)m * RP + n] = res;  // zero-padded cols >= 150
  }
}

// ---------------------------------------------------------------------------
// Kernel 2: recurrence. 16 WGs x 8 waves; each WG owns 16 batch rows for all
// 512 steps. Packed weights stream from L2; h (double-buffered) + P in LDS.
// ---------------------------------------------------------------------------
__global__ void __launch_bounds__(256)
farnn_recurrence(const float* __restrict__ Lall,   // [B*L, RP]
                 const v2f*   __restrict__ T1p,    // packed [10][50][32]
                 const v2f*   __restrict__ T2tp,   // packed [13][40][32] (T2^T)
                 const v2f*   __restrict__ Wp,     // packed [13][50][32]
                 float* __restrict__ out)          // [B, L, SAS]
{
  __shared__ float hbuf[2][16][HP];
  __shared__ float P[16][PP];

  const int lane = threadIdx.x & 31;
  const int wv   = threadIdx.x >> 5;
  const int half = lane >> 4;
  const int l15  = lane & 15;
  const int batchBase = blockIdx.x * 16;

  // h0 = one-hot state 0 (cols 0..199 used; padding never read)
  for (int i = threadIdx.x; i < 16 * SAS; i += 256) {
    const int r = i / SAS, c = i % SAS;
    hbuf[0][r][c] = (c == 0) ? 1.0f : 0.0f;
  }
  __syncthreads();

  for (int t = 0; t < L_SEQ; ++t) {
    const int cur = t & 1;
    const int nxt = cur ^ 1;

    // ---- Stage A: P = Lt .* (h @ T1) over 10 R-tiles ----
    for (int nt = wv; nt < T1_NT; nt += 8) {       // wave-uniform
      const int nn = nt * 16 + l15;
      const v2f* bp = T1p + (size_t)nt * T1_KS * 32 + lane;
      v8f acc = {};
      for (int ks = 0; ks < T1_KS; ++ks) {
        v2f a = *(const v2f*)&hbuf[cur][l15][4 * ks + 2 * half];
        v2f b = bp[ks * 32];
        acc = wmma_f32_k4(a, b, acc);
      }
#pragma unroll
      for (int v = 0; v < 8; ++v) {
        const int m = v + 8 * half;
        const float lt = Lall[((long)(batchBase + m) * L_SEQ + t) * RP + nn];
        P[m][nn] = lt * acc[v];                    // zero for nn >= 150
      }
    }
    __syncthreads();

    // ---- Stage B: h' = relu(P @ T2^T + h @ W) over 13 SAS-tiles ----
    for (int nt = wv; nt < T2_NT; nt += 8) {       // wave-uniform
      const int nn = nt * 16 + l15;
      const v2f* bp2 = T2tp + (size_t)nt * T2_KS * 32 + lane;
      const v2f* bpw = Wp   + (size_t)nt * W_KS  * 32 + lane;
      v8f acc = {};
      for (int ks = 0; ks < T2_KS; ++ks) {         // language term, K=160
        v2f a = *(const v2f*)&P[l15][4 * ks + 2 * half];
        v2f b = bp2[ks * 32];
        acc = wmma_f32_k4(a, b, acc);
      }
      for (int ks = 0; ks < W_KS; ++ks) {          // wildcard term, K=200
        v2f a = *(const v2f*)&hbuf[cur][l15][4 * ks + 2 * half];
        v2f b = bpw[ks * 32];
        acc = wmma_f32_k4(a, b, acc);
      }
#pragma unroll
      for (int v = 0; v < 8; ++v) {
        const int m = v + 8 * half;
        const float val = fmaxf(acc[v], 0.0f);
        hbuf[nxt][m][nn] = val;
        if (nn < SAS)
          out[((long)(batchBase + m) * L_SEQ + t) * SAS + nn] = val;
      }
    }
    __syncthreads();
  }
}

extern "C" void kernel_launch(void* const* d_in, const int* in_sizes, int n_in,
                              void* d_out, int out_size, void* d_ws, size_t ws_size,
                              hipStream_t stream) {
  const int*   input     = (const int*)d_in[0];
  // d_in[1] = lengths (unused by the reference output)
  const float* embedding = (const float*)d_in[2];
  const float* embed_r   = (const float*)d_in[3];
  const float* G         = (const float*)d_in[4];
  const float* T1        = (const float*)d_in[5];
  const float* T2        = (const float*)d_in[6];
  const float* W         = (const float*)d_in[7];
  const float* beta      = (const float*)d_in[8];
  float* out = (float*)d_out;

  // workspace layout
  float* Lall = (float*)d_ws;                         // B*L*RP floats (84 MB)
  size_t off  = (size_t)B_SZ * L_SEQ * RP;
  v2f* Gp   = (v2f*)(Lall + off);  off += (size_t)G_NT  * G_KS  * 32 * 2;
  v2f* T1p  = (v2f*)(Lall + off);  off += (size_t)T1_NT * T1_KS * 32 * 2;
  v2f* T2tp = (v2f*)(Lall + off);  off += (size_t)T2_NT * T2_KS * 32 * 2;
  v2f* Wp   = (v2f*)(Lall + off);

  // pack B operands (zero-padded, WMMA lane order)
  const int gTot  = G_NT  * G_KS  * 32;
  const int t1Tot = T1_NT * T1_KS * 32;
  const int t2Tot = T2_NT * T2_KS * 32;
  const int wTot  = W_NT  * W_KS  * 32;
  pack_b<<<(gTot  + 255) / 256, 256, 0, stream>>>(G,  D_DIM, R_DIM, R_DIM, 1, G_KS,  gTot,  Gp);
  pack_b<<<(t1Tot + 255) / 256, 256, 0, stream>>>(T1, SAS,   R_DIM, R_DIM, 1, T1_KS, t1Tot, T1p);
  // T2^T: B[k=r][n=s] = T2[s][r] -> strideK=1, strideN=R
  pack_b<<<(t2Tot + 255) / 256, 256, 0, stream>>>(T2, R_DIM, SAS, 1, R_DIM, T2_KS, t2Tot, T2tp);
  pack_b<<<(wTot  + 255) / 256, 256, 0, stream>>>(W,  SAS,   SAS, SAS,   1, W_KS,  wTot,  Wp);

  const int rows  = B_SZ * L_SEQ;                 // 131072
  const int tiles = (rows / 16) * G_NT;           // 81920 waves
  farnn_embed_mix<<<tiles / 8, 256, 0, stream>>>(input, embedding, embed_r,
                                                 Gp, beta, Lall);
  farnn_recurrence<<<B_SZ / 16, 256, 0, stream>>>(Lall, T1p, T2tp, Wp, out);
}